// PdfSampler_88364657148061
// MI455X (gfx1250) — compile-verified
//
#include <hip/hip_runtime.h>

#define NS   64      // coarse samples
#define NF   128     // fine samples
#define NZ   192     // combined
#define RPW  16      // rays per wave
#define WPB  4       // waves per block
#define RPB  64      // rays per block

typedef float        v2f __attribute__((ext_vector_type(2)));
typedef float        v8f __attribute__((ext_vector_type(8)));
typedef unsigned int v4u __attribute__((ext_vector_type(4)));
typedef int          v8i __attribute__((ext_vector_type(8)));
typedef int          v4i __attribute__((ext_vector_type(4)));

__global__ __launch_bounds__(128, 1)
void nerf_pdf_kernel(const float* __restrict__ nearp,
                     const float* __restrict__ farp,
                     const float* __restrict__ dens,
                     const float* __restrict__ rgb,
                     float* __restrict__ out, int B)
{
  __shared__ float sDens[RPB][NS];   // 16 KB  (TDM destination)
  __shared__ float sCdf [RPB][NS];   // 16 KB
  __shared__ float sFine[RPB][NF];   // 32 KB
  __shared__ float sNear[RPB];
  __shared__ float sDz  [RPB];

  const int lane     = threadIdx.x & 31;
  const int wave     = threadIdx.x >> 5;
  const int rW       = wave * RPW;                 // wave's ray slice in LDS
  const int rayBaseW = blockIdx.x * RPB + rW;

  float* outZ   = out;
  float* outRGB = out + (size_t)B * NZ;
  float* outD   = outRGB + (size_t)B * 3;
  float* outA   = outD + B;

  // ---- TDM: async DMA this wave's 16x64 fp32 density tile into LDS ----
#if __has_builtin(__builtin_amdgcn_tensor_load_to_lds)
  {
    unsigned long long ga = (unsigned long long)(size_t)(dens + (size_t)rayBaseW * NS);
    unsigned la = (unsigned)(size_t)(&sDens[rW][0]);
    v4u g0 = { 1u,                                            // count=1 (valid D#)
               la,                                            // lds_addr
               (unsigned)(ga & 0xffffffffu),                  // global_addr lo
               (unsigned)((ga >> 32) & 0x1ffffffu) | (2u << 30) }; // addr hi | type=2
    v8i g1 = { (int)(2u << 16),                               // data_size = 4B
               (int)((RPW * NS) << 16),                       // tensor_dim0 lo16 = 1024
               (int)(((RPW * NS) >> 16) & 0xffff) | (1 << 16),// dim0 hi | tensor_dim1 = 1
               (int)((RPW * NS) << 16),                       // tile_dim0 = 1024 (1-D tile)
               0,                                             // tile_dim1 = 0, tile_dim2 = 0
               (int)(RPW * NS),                               // tensor_dim0_stride
               0, 0 };
    v4i gz4 = { 0, 0, 0, 0 };
    v8i gz8 = { 0, 0, 0, 0, 0, 0, 0, 0 };
    __builtin_amdgcn_tensor_load_to_lds(g0, g1, gz4, gz4, gz8, 0);
  }
#else
  for (int i = lane; i < RPW * NS; i += 32)
    sDens[rW + (i >> 6)][i & 63] = dens[(size_t)rayBaseW * NS + i];
#endif

  // ---- prefetch the block's rgb stream (read-once, 48 KB) while DMA runs ----
  {
    const char* base = (const char*)(rgb + (size_t)(blockIdx.x * RPB) * NS * 3);
    const int bytes = RPB * NS * 3 * 4;
    for (int off = threadIdx.x * 256; off < bytes; off += 128 * 256)
      __builtin_prefetch(base + off, 0, 1);
  }

  // ---- per-ray scalars (uniform bins: delta is a per-ray constant) ----
  if (lane < RPW) {
    int r = rayBaseW + lane;
    float n = nearp[r], f = farp[r];
    sNear[rW + lane] = n;
    sDz  [rW + lane] = (f - n) * (1.0f / NS);
  }

#if __has_builtin(__builtin_amdgcn_s_wait_tensorcnt)
  __builtin_amdgcn_s_wait_tensorcnt(0);
#endif
  __asm__ volatile("" ::: "memory");

  // ---- build A fragments: s[k] = density*dz, 16 rays x 64 K, f32 16x16x4 layout ----
  const int half = lane >> 4;
  const int mlow = lane & 15;

  const float dzA = sDz[rW + mlow];
  v2f a[16];
  {
    const float* dr = &sDens[rW + mlow][0];
#pragma unroll
    for (int c = 0; c < 16; ++c) {
      v2f d = *(const v2f*)(dr + 4 * c + 2 * half);
      a[c] = d * dzA;
    }
  }

  // B fragments: constant triangular "<=" matrix, K-chunked
  v2f bOne = { 1.0f, 1.0f };
  v2f bDiag[4];
#pragma unroll
  for (int d = 0; d < 4; ++d) {
    bDiag[d].x = (4 * d + 2 * half)     <= mlow ? 1.0f : 0.0f;
    bDiag[d].y = (4 * d + 2 * half + 1) <= mlow ? 1.0f : 0.0f;
  }

  // ---- matrix-engine inclusive prefix sum: I = S x U (40 wmma / 16 rays) ----
  v8f SI[4];
#pragma unroll
  for (int t = 0; t < 4; ++t) {
    v8f c = { 0, 0, 0, 0, 0, 0, 0, 0 };
#pragma unroll
    for (int k = 0; k <= 4 * t + 3; ++k) {
      v2f b = (k < 4 * t) ? bOne : bDiag[k - 4 * t];
      c = __builtin_amdgcn_wmma_f32_16x16x4_f32(false, a[k], false, b,
                                                (short)0, c, false, false);
    }
    SI[t] = c;
  }

  // ---- weights / cdf (telescoped) / rgb-depth-acc reductions ----
#pragma unroll
  for (int g = 0; g < 8; ++g) {
    const int M    = g + 8 * half;          // ray within wave for C/D layout
    const int rayG = rayBaseW + M;
    const float nM  = sNear[rW + M];
    const float dzM = sDz  [rW + M];
    const float tot  = __shfl(SI[3][g], 15, 16);            // I_63 broadcast per half
    const float invC = 1.0f / ((1.0f - __expf(-tot)) + 1e-6f);

    float wsum = 0.f, dsum = 0.f, rs = 0.f, gs = 0.f, bs = 0.f;
#pragma unroll
    for (int t = 0; t < 4; ++t) {
      const int j    = 16 * t + mlow;
      const float si = SI[t][g];
      const float sv = sDens[rW + M][j] * dzM;
      const float eI = __expf(-si);
      const float w  = __expf(sv - si) - eI;                // alpha * transmittance
      sCdf[rW + M][j] = (1.0f - eI) * invC;                 // cumsum(pdf)
      const float mid = fmaf(dzM, (float)j + 0.5f, nM);
      const float* rp = rgb + ((size_t)rayG * NS + j) * 3;
      wsum += w; dsum += w * mid;
      rs += w * rp[0]; gs += w * rp[1]; bs += w * rp[2];
    }
#pragma unroll
    for (int mask = 8; mask >= 1; mask >>= 1) {
      wsum += __shfl_xor(wsum, mask, 16);
      dsum += __shfl_xor(dsum, mask, 16);
      rs   += __shfl_xor(rs,   mask, 16);
      gs   += __shfl_xor(gs,   mask, 16);
      bs   += __shfl_xor(bs,   mask, 16);
    }
    if (mlow == 0) {
      outRGB[(size_t)rayG * 3 + 0] = rs;
      outRGB[(size_t)rayG * 3 + 1] = gs;
      outRGB[(size_t)rayG * 3 + 2] = bs;
      outD[rayG] = dsum / (wsum + 1e-8f);
      outA[rayG] = wsum;
    }
  }

  // ---- fine sampling + rank-merge scatter (no sort needed) ----
#pragma unroll 1
  for (int m = 0; m < RPW; ++m) {
    const int rayG   = rayBaseW + m;
    const float nM   = sNear[rW + m];
    const float dzM  = sDz  [rW + m];
    const float* cdfR = &sCdf[rW + m][0];
    float* fineR      = &sFine[rW + m][0];
    float* zRow       = outZ + (size_t)rayG * NZ;

#pragma unroll
    for (int q = 0; q < 4; ++q) {
      const int k = lane + 32 * q;
      const float u = 0.05f + (float)k * (0.9f / 127.0f);
      int cnt = 0;                                  // searchsorted-right over cdf
#pragma unroll
      for (int s = 32; s >= 1; s >>= 1) {
        int c2 = cnt + s;
        if (cdfR[c2 - 1] <= u) cnt = c2;
      }
      const int below = cnt > 0 ? cnt - 1 : 0;
      const int above = cnt < NS ? cnt : NS - 1;
      const float cb = cdfR[below];
      const float ca = cdfR[above];
      const float mb = fmaf(dzM, (float)below + 0.5f, nM);
      const float ma = fmaf(dzM, (float)above + 0.5f, nM);
      float den = ca - cb;
      den = den < 1e-5f ? 1.0f : den;
      float tt = (u - cb) / den;
      tt = fminf(fmaxf(tt, 0.0f), 1.0f);            // keeps fine monotone under fp
      const float fv = fmaf(tt, ma - mb, mb);
      fineR[k] = fv;
      int cm = 0;                                   // #mids <= fv (same fmaf expr)
#pragma unroll
      for (int s = 32; s >= 1; s >>= 1) {
        int c2 = cm + s;
        if (fmaf(dzM, (float)(c2 - 1) + 0.5f, nM) <= fv) cm = c2;
      }
      zRow[k + cm] = fv;                            // fine rank = k + #mids<=fv
    }

#pragma unroll
    for (int q = 0; q < 2; ++q) {
      const int j = lane + 32 * q;
      const float y = fmaf(dzM, (float)j + 0.5f, nM);
      int cf = 0;                                   // #fine < y (strict: complement)
#pragma unroll
      for (int s = 64; s >= 1; s >>= 1) {
        int c2 = cf + s;
        if (fineR[c2 - 1] < y) cf = c2;
      }
      zRow[j + cf] = y;                             // mid rank = j + #fine<y
    }
  }
}

extern "C" void kernel_launch(void* const* d_in, const int* in_sizes, int n_in,
                              void* d_out, int out_size, void* d_ws, size_t ws_size,
                              hipStream_t stream) {
  const float* nearp = (const float*)d_in[0];
  const float* farp  = (const float*)d_in[1];
  const float* dens  = (const float*)d_in[2];
  const float* rgb   = (const float*)d_in[3];
  const int B = in_sizes[0];                 // 131072
  dim3 grid(B / RPB), block(WPB * 32);
  hipLaunchKernelGGL(nerf_pdf_kernel, grid, block, 0, stream,
                     nearp, farp, dens, rgb, (float*)d_out, B);
}